// RayMaskedBEVAttention_32160715112570
// MI455X (gfx1250) — compile-verified
//
#include <hip/hip_runtime.h>
#include <math.h>

// ---------------------------------------------------------------------------
// RayMaskedBEVAttention for MI455X (gfx1250, wave32).
//
// Strategy (roofline): K/V projections are precomputed per BEV cell (16384
// cells) instead of per (query,candidate) -> total FLOPs drop from ~43 GF to
// ~2 GF. K_all/V_all (16.8 MB fp32) are L2-resident (192 MB L2), so the
// 1 GB candidate gather runs at L2 bandwidth. Dense projections use
// v_wmma_f32_16x16x4_f32 (full fp32 precision, CDNA5 WMMA path).
// ---------------------------------------------------------------------------

typedef float v2f __attribute__((ext_vector_type(2)));
typedef float v8f __attribute__((ext_vector_type(8)));

constexpr int NRQ = 4096;   // Nr = 16*256 rays (queries)
constexpr int NBC = 16384;  // Nb = 128*128 BEV cells
constexpr int CB  = 80;     // BEV channels
constexpr int CRV = 384;    // ray-view channels
constexpr int DD  = 128;    // head dim
constexpr int MM  = 256;    // candidates per query
constexpr float INV_SQRT_D = 0.08838834764831845f; // 1/sqrt(128)

// WMMA fragment helpers (CDNA5 layouts, wave32):
//   A 16x4 f32 : lane l in [0,16): holds A[l][koff], A[l][koff+1], koff = 2*(l>>4)
//   B 4x16 f32 : lane l: col = l&15, holds B[koff][col], B[koff+1][col]
//   C/D 16x16  : VGPR r -> row r + 8*(lane>>4), col = lane&15

// ---------------------------------------------------------------------------
// Kernel 1: Q = l2norm(rv_flat @ Wq + bq)   (4096x384)@(384x128)
// Block = 256 threads = 8 waves; each wave one 16x16 tile; block covers a
// full 16-row x 128-col slab so the row l2-norm reduces across the 8 waves.
// ---------------------------------------------------------------------------
__global__ __launch_bounds__(256)
void rmba_qproj_kernel(const float* __restrict__ A,    // (NRQ, CRV) row-major
                       const float* __restrict__ Wq,   // (CRV, DD) row-major
                       const float* __restrict__ bq,   // (DD)
                       float* __restrict__ Qn)         // (NRQ, DD)
{
    const int lane = threadIdx.x & 31;
    const int wave = threadIdx.x >> 5;
    const int half = lane >> 4;
    const int l16  = lane & 15;
    const int koff = half * 2;
    const int rowBase = blockIdx.x * 16;
    const int colBase = wave * 16;
    const int arow = rowBase + l16;
    const int bcol = colBase + l16;

    v8f acc = {};
    #pragma unroll 4
    for (int k = 0; k < CRV; k += 4) {
        v2f a, b;
        a.x = A[arow * CRV + k + koff];
        a.y = A[arow * CRV + k + koff + 1];
        b.x = Wq[(k + koff) * DD + bcol];
        b.y = Wq[(k + koff + 1) * DD + bcol];
        acc = __builtin_amdgcn_wmma_f32_16x16x4_f32(false, a, false, b,
                                                    (short)0, acc, false, false);
    }

    const float bias = bq[bcol];
    float vals[8], ss[8];
    #pragma unroll
    for (int r = 0; r < 8; ++r) { vals[r] = acc[r] + bias; ss[r] = vals[r] * vals[r]; }

    // sum across the 16 lanes of this half (xor masks 1..8 stay in-half)
    #pragma unroll
    for (int s = 8; s >= 1; s >>= 1)
        #pragma unroll
        for (int r = 0; r < 8; ++r) ss[r] += __shfl_xor(ss[r], s, 32);

    __shared__ float sq[8][16];
    __shared__ float rnorm[16];
    if (l16 == 0) {
        #pragma unroll
        for (int r = 0; r < 8; ++r) sq[wave][half * 8 + r] = ss[r];
    }
    __syncthreads();
    if (threadIdx.x < 16) {
        float s = 0.f;
        #pragma unroll
        for (int w = 0; w < 8; ++w) s += sq[w][threadIdx.x];
        rnorm[threadIdx.x] = 1.0f / fmaxf(sqrtf(s), 1e-12f);
    }
    __syncthreads();
    #pragma unroll
    for (int r = 0; r < 8; ++r) {
        const int row = half * 8 + r;
        Qn[(rowBase + row) * DD + bcol] = vals[r] * rnorm[row];
    }
}

// ---------------------------------------------------------------------------
// Kernel 2: K_all = l2norm(bev @ Wk + bk), V_all = bev @ Wv + bv, for all
// 16384 cells. bev is physically (CB, NBC): A is accessed transposed.
// Two accumulators per wave share the A fragment.
// ---------------------------------------------------------------------------
__global__ __launch_bounds__(256)
void rmba_kvproj_kernel(const float* __restrict__ bev,  // (CB, NBC)
                        const float* __restrict__ Wk, const float* __restrict__ bk,
                        const float* __restrict__ Wv, const float* __restrict__ bv,
                        float* __restrict__ Kall,       // (NBC, DD)
                        float* __restrict__ Vall)       // (NBC, DD)
{
    const int lane = threadIdx.x & 31;
    const int wave = threadIdx.x >> 5;
    const int half = lane >> 4;
    const int l16  = lane & 15;
    const int koff = half * 2;
    const int rowBase = blockIdx.x * 16;   // BEV cell index base
    const int colBase = wave * 16;
    const int arow = rowBase + l16;
    const int bcol = colBase + l16;

    v8f acck = {}, accv = {};
    #pragma unroll 4
    for (int k = 0; k < CB; k += 4) {
        v2f a, wk, wv;
        a.x  = bev[(k + koff) * NBC + arow];      // transposed A read
        a.y  = bev[(k + koff + 1) * NBC + arow];
        wk.x = Wk[(k + koff) * DD + bcol];
        wk.y = Wk[(k + koff + 1) * DD + bcol];
        wv.x = Wv[(k + koff) * DD + bcol];
        wv.y = Wv[(k + koff + 1) * DD + bcol];
        acck = __builtin_amdgcn_wmma_f32_16x16x4_f32(false, a, false, wk,
                                                     (short)0, acck, false, false);
        accv = __builtin_amdgcn_wmma_f32_16x16x4_f32(false, a, false, wv,
                                                     (short)0, accv, false, false);
    }

    const float biask = bk[bcol];
    const float biasv = bv[bcol];
    float kv[8], ss[8];
    #pragma unroll
    for (int r = 0; r < 8; ++r) { kv[r] = acck[r] + biask; ss[r] = kv[r] * kv[r]; }
    #pragma unroll
    for (int s = 8; s >= 1; s >>= 1)
        #pragma unroll
        for (int r = 0; r < 8; ++r) ss[r] += __shfl_xor(ss[r], s, 32);

    __shared__ float sq[8][16];
    __shared__ float rnorm[16];
    if (l16 == 0) {
        #pragma unroll
        for (int r = 0; r < 8; ++r) sq[wave][half * 8 + r] = ss[r];
    }
    __syncthreads();
    if (threadIdx.x < 16) {
        float s = 0.f;
        #pragma unroll
        for (int w = 0; w < 8; ++w) s += sq[w][threadIdx.x];
        rnorm[threadIdx.x] = 1.0f / fmaxf(sqrtf(s), 1e-12f);
    }
    __syncthreads();
    #pragma unroll
    for (int r = 0; r < 8; ++r) {
        const int row = half * 8 + r;
        Kall[(rowBase + row) * DD + bcol] = kv[r] * rnorm[row];
        Vall[(rowBase + row) * DD + bcol] = accv[r] + biasv;
    }
}

// ---------------------------------------------------------------------------
// Kernel 3: per-query masked attention over 256 gathered cells (L2-resident
// K_all/V_all). One block (256 thr / 8 waves) per query; one candidate per
// thread for logits+softmax, then a (d,m-half) layout for the attn@V pass
// so V rows are read coalesced.
// ---------------------------------------------------------------------------
__global__ __launch_bounds__(256)
void rmba_attn_kernel(const float* __restrict__ Qn,    // (NRQ, DD)
                      const float* __restrict__ Kall,  // (NBC, DD)
                      const float* __restrict__ Vall,  // (NBC, DD)
                      const int*   __restrict__ cand,  // (NRQ, MM)
                      float* __restrict__ attn_out,    // (NRQ, MM)
                      float* __restrict__ ctx)         // (NRQ, DD)
{
    const int n   = blockIdx.x;
    const int tid = threadIdx.x;

    __shared__ float q_sm[DD];
    __shared__ float a_sm[MM];
    __shared__ int   idx_sm[MM];
    __shared__ float red_sm[8];
    __shared__ float cpart[256];

    if (tid < DD) q_sm[tid] = Qn[n * DD + tid];
    const int idx = cand[n * MM + tid];
    idx_sm[tid] = idx;
    __syncthreads();

    // logits: one candidate per thread; each lane streams its 512B K row
    float logit = -INFINITY;
    if (idx >= 0) {
        const float4* krow = (const float4*)(Kall + (size_t)idx * DD);
        const float4* qv   = (const float4*)q_sm;
        float acc = 0.f;
        #pragma unroll
        for (int c = 0; c < DD / 4; ++c) {
            const float4 k4 = krow[c];
            const float4 q4 = qv[c];
            acc += k4.x * q4.x + k4.y * q4.y + k4.z * q4.z + k4.w * q4.w;
        }
        logit = acc * INV_SQRT_D;
    }

    // block max (wave32 shuffle tree + LDS across the 8 waves)
    float mx = logit;
    #pragma unroll
    for (int s = 16; s >= 1; s >>= 1) mx = fmaxf(mx, __shfl_xor(mx, s, 32));
    if ((tid & 31) == 0) red_sm[tid >> 5] = mx;
    __syncthreads();
    float bm = red_sm[0];
    #pragma unroll
    for (int w = 1; w < 8; ++w) bm = fmaxf(bm, red_sm[w]);
    __syncthreads();

    const float e = (idx >= 0) ? __expf(logit - bm) : 0.f;
    float sm = e;
    #pragma unroll
    for (int s = 16; s >= 1; s >>= 1) sm += __shfl_xor(sm, s, 32);
    if ((tid & 31) == 0) red_sm[tid >> 5] = sm;
    __syncthreads();
    float bs = 0.f;
    #pragma unroll
    for (int w = 0; w < 8; ++w) bs += red_sm[w];

    const float a = e / bs;               // exactly 0 at padded slots
    attn_out[n * MM + tid] = a;
    a_sm[tid] = a;
    __syncthreads();

    // ctx[d] = sum_m a[m] * V[idx_m][d]; threads 0-127 take m in [0,128),
    // threads 128-255 take m in [128,256) -> V rows read fully coalesced.
    const int d     = tid & 127;
    const int mBase = (tid >> 7) * 128;
    float c = 0.f;
    for (int m = 0; m < 128; ++m) {
        const int ix = idx_sm[mBase + m];
        if (ix >= 0) c += a_sm[mBase + m] * Vall[(size_t)ix * DD + d];
    }
    cpart[tid] = c;
    __syncthreads();
    if (tid < DD) ctx[n * DD + tid] = cpart[tid] + cpart[tid + 128];
}

// ---------------------------------------------------------------------------
// Kernel 4: out = ctx @ Wo + bo   (4096x128)@(128x384) -> d_out[0:NRQ*CRV]
// ---------------------------------------------------------------------------
__global__ __launch_bounds__(256)
void rmba_oproj_kernel(const float* __restrict__ ctx,  // (NRQ, DD)
                       const float* __restrict__ Wo,   // (DD, CRV)
                       const float* __restrict__ bo,   // (CRV)
                       float* __restrict__ out)        // (NRQ, CRV)
{
    const int mTiles = NRQ / 16;          // 256
    const int mTile  = blockIdx.x % mTiles;
    const int nGrp   = blockIdx.x / mTiles;  // 0..2, each = 128 cols
    const int lane = threadIdx.x & 31;
    const int wave = threadIdx.x >> 5;
    const int half = lane >> 4;
    const int l16  = lane & 15;
    const int koff = half * 2;
    const int rowBase = mTile * 16;
    const int colBase = nGrp * 128 + wave * 16;
    const int arow = rowBase + l16;
    const int bcol = colBase + l16;

    v8f acc = {};
    #pragma unroll 4
    for (int k = 0; k < DD; k += 4) {
        v2f a, b;
        a.x = ctx[arow * DD + k + koff];
        a.y = ctx[arow * DD + k + koff + 1];
        b.x = Wo[(k + koff) * CRV + bcol];
        b.y = Wo[(k + koff + 1) * CRV + bcol];
        acc = __builtin_amdgcn_wmma_f32_16x16x4_f32(false, a, false, b,
                                                    (short)0, acc, false, false);
    }
    const float bias = bo[bcol];
    #pragma unroll
    for (int r = 0; r < 8; ++r) {
        const int row = rowBase + half * 8 + r;
        out[row * CRV + bcol] = acc[r] + bias;
    }
}

// ---------------------------------------------------------------------------
// Launch: rv_x, bev_feat, cand_mat, Wq, bq, Wk, bk, Wv, bv, Wo, bo
// d_out = [ out (NRQ*CRV) | attn (NRQ*MM) ]
// ---------------------------------------------------------------------------
extern "C" void kernel_launch(void* const* d_in, const int* in_sizes, int n_in,
                              void* d_out, int out_size, void* d_ws, size_t ws_size,
                              hipStream_t stream) {
    const float* rv_x = (const float*)d_in[0];   // (1,16,256,384) = (NRQ,CRV)
    const float* bev  = (const float*)d_in[1];   // (1,80,128,128) = (CB,NBC)
    const int*   cand = (const int*)  d_in[2];   // (NRQ, MM)
    const float* Wq   = (const float*)d_in[3];
    const float* bq   = (const float*)d_in[4];
    const float* Wk   = (const float*)d_in[5];
    const float* bk   = (const float*)d_in[6];
    const float* Wv   = (const float*)d_in[7];
    const float* bv   = (const float*)d_in[8];
    const float* Wo   = (const float*)d_in[9];
    const float* bo   = (const float*)d_in[10];

    float* out_main = (float*)d_out;                       // NRQ*CRV
    float* out_attn = out_main + (size_t)NRQ * CRV;        // NRQ*MM

    // workspace: Qn | Kall | Vall | ctx  (~20 MB total)
    float* Qn   = (float*)d_ws;
    float* Kall = Qn   + (size_t)NRQ * DD;
    float* Vall = Kall + (size_t)NBC * DD;
    float* ctx  = Vall + (size_t)NBC * DD;

    rmba_qproj_kernel <<<NRQ / 16, 256, 0, stream>>>(rv_x, Wq, bq, Qn);
    rmba_kvproj_kernel<<<NBC / 16, 256, 0, stream>>>(bev, Wk, bk, Wv, bv, Kall, Vall);
    rmba_attn_kernel  <<<NRQ,      256, 0, stream>>>(Qn, Kall, Vall, cand, out_attn, ctx);
    rmba_oproj_kernel <<<(NRQ / 16) * (CRV / 128), 256, 0, stream>>>(ctx, Wo, bo, out_main);
}